// BaseModel_2654289789315
// MI455X (gfx1250) — compile-verified
//
#include <hip/hip_runtime.h>
#include <math.h>

#define N_ATOMS  10000
#define N_PAIRS  160000
#define N_CH     32
#define KDIM     128
#define K0TOT    384

typedef float v2f __attribute__((ext_vector_type(2)));
typedef float v8f __attribute__((ext_vector_type(8)));

// U2 (9x9) from the reference, hardcoded. a3=1/sqrt3, a2=1/sqrt2, b2=2/sqrt6, b1=1/sqrt6
#define A3 0.5773502691896258f
#define A2 0.7071067811865476f
#define B2 0.8164965809277260f
#define B1 0.4082482904638630f
__constant__ float U2c[9][9] = {
    { A3, 0,   0,   0,  A3, 0,   0,   0,   A3}, // row0
    { 0,  0,   0,   0,  0,  A2,  0,  -A2,  0 }, // row1
    { 0,  0,  -A2,  0,  0,  0,   A2,  0,   0 }, // row2
    { 0,  A2,  0,  -A2, 0,  0,   0,   0,   0 }, // row3
    { 0,  0,   A2,  0,  0,  0,   A2,  0,   0 }, // row4
    { 0,  A2,  0,   A2, 0,  0,   0,   0,   0 }, // row5
    {-B1, 0,   0,   0,  B2, 0,   0,   0,  -B1}, // row6
    { 0,  0,   0,   0,  0,  A2,  0,   A2,  0 }, // row7
    {-A2, 0,   0,   0,  0,  0,   0,   0,   A2}, // row8
};

__global__ void k_zero(float* p, long n) {
    long i = (long)blockIdx.x * blockDim.x + threadIdx.x;
    if (i < n) p[i] = 0.f;
}

__global__ void k_init_out(float* out, const float* b_out) {
    int i = blockIdx.x * blockDim.x + threadIdx.x;
    if (i < N_ATOMS) out[i] = b_out[0];
}

// one block (128 threads) per pair; thread k handles channel k
__global__ void k_pairs(const float* __restrict__ pos, const int* __restrict__ species,
                        const int* __restrict__ pairs, const float* __restrict__ emb,
                        const float* __restrict__ W_rad,
                        float* __restrict__ f0, float* __restrict__ f1, float* __restrict__ f2) {
    int p = blockIdx.x;
    int k = threadIdx.x;                     // 0..127
    int c  = pairs[2 * p + 0];
    int nb = pairs[2 * p + 1];
    float rx = pos[3 * nb + 0] - pos[3 * c + 0];
    float ry = pos[3 * nb + 1] - pos[3 * c + 1];
    float rz = pos[3 * nb + 2] - pos[3 * c + 2];
    float d  = sqrtf(rx * rx + ry * ry + rz * rz + 1e-12f);
    float inv = 1.f / d;
    float ux = rx * inv, uy = ry * inv, uz = rz * inv;

    __shared__ float s_rb[8];
    if (k < 8) {
        float mu = 20.f * (float)k / 7.f;          // linspace(0,20,8)
        float t  = (d - 15.f) * 0.2f;              // (d-(20-5))/5
        t = fminf(fmaxf(t, 0.f), 1.f);
        float fc = 0.5f * (cosf(3.14159265358979f * t) + 1.f);
        float x  = (d - mu) * 0.4f;                // sigma = 2.5
        s_rb[k] = expf(-x * x) * fc;
    }
    __syncthreads();

    float neigh = emb[species[nb] * N_CH + (k >> 2)];  // h0 layout: ch*4+m
    float R[3];
#pragma unroll
    for (int l = 0; l < 3; ++l) {
        float s = 0.f;
#pragma unroll
        for (int j = 0; j < 8; ++j) s += s_rb[j] * W_rad[(l * 8 + j) * KDIM + k];
        R[l] = s;
    }
    const float mp = 0.1f;                        // MP_SCALING
    atomicAdd(&f0[(long)c * KDIM + k], R[0] * neigh * mp);
    float sh1[3] = { uy, uz, ux };
    float w1 = R[1] * neigh * mp;
#pragma unroll
    for (int m = 0; m < 3; ++m)
        atomicAdd(&f1[((long)c * 3 + m) * KDIM + k], sh1[m] * w1);
    const float s3 = 1.7320508075688772f;
    float sh2[5] = { s3 * ux * uy, s3 * uy * uz, 0.5f * (3.f * uz * uz - 1.f),
                     s3 * ux * uz, 0.5f * s3 * (ux * ux - uy * uy) };
    float w2 = R[2] * neigh * mp;
#pragma unroll
    for (int m = 0; m < 5; ++m)
        atomicAdd(&f2[((long)c * 5 + m) * KDIM + k], sh2[m] * w2);
}

// f_unc[n,p,k] for l=1 and l=2 (U2^T applied to padded rows)
__global__ void k_uncouple(const float* __restrict__ f1, const float* __restrict__ f2,
                           float* __restrict__ u1, float* __restrict__ u2) {
    long t = (long)blockIdx.x * blockDim.x + threadIdx.x;
    if (t >= (long)N_ATOMS * 9 * KDIM) return;
    int k = (int)(t % KDIM);
    int p = (int)((t / KDIM) % 9);
    long n = t / (9 * KDIM);
    float a = 0.f, b = 0.f;
#pragma unroll
    for (int q = 0; q < 3; ++q) a += U2c[1 + q][p] * f1[(n * 3 + q) * KDIM + k];
#pragma unroll
    for (int q = 0; q < 5; ++q) b += U2c[4 + q][p] * f2[(n * 5 + q) * KDIM + k];
    u1[t] = a;
    u2[t] = b;
}

// fp32 WMMA GEMM: C[M x 128] = A[M x 128] * B[128 x 128].
// One wave computes a full 16x128 row-stripe: the A fragment is loaded once per
// K-step and reused across 8 accumulator tiles (8 back-to-back v_wmma).
// grid.x = M/16, block = 32 (EXEC all ones, exact tiles only).
__global__ void k_gemm_wmma128(const float* __restrict__ A, const float* __restrict__ B,
                               float* __restrict__ C) {
    int m0 = blockIdx.x * 16;
    int lane = threadIdx.x;
    int half = lane >> 4;        // 0: K{0,1}/rows 0-7, 1: K{2,3}/rows 8-15
    int l16  = lane & 15;
    const float* Arow = A + (long)(m0 + l16) * KDIM;
    v8f acc[8] = {};
    for (int kk = 0; kk < KDIM; kk += 4) {
        int ka = kk + 2 * half;
        v2f a;
        a.x = Arow[ka];
        a.y = Arow[ka + 1];
        const float* B0 = B + (long)ka * KDIM + l16;
#pragma unroll
        for (int t = 0; t < 8; ++t) {
            v2f b;
            b.x = B0[t * 16];
            b.y = B0[KDIM + t * 16];
            acc[t] = __builtin_amdgcn_wmma_f32_16x16x4_f32(false, a, false, b, (short)0,
                                                           acc[t], false, false);
        }
    }
#pragma unroll
    for (int t = 0; t < 8; ++t) {
        float* Cc = C + (long)(m0 + 8 * half) * KDIM + t * 16 + l16;
#pragma unroll
        for (int i = 0; i < 8; ++i) Cc[(long)i * KDIM] = acc[t][i];
    }
}

// feats + couple(row0 only = trace/sqrt3) + concat + center-embedding -> x0 (N x 384)
__global__ void k_feats(const float* __restrict__ f0, const float* __restrict__ u1,
                        const float* __restrict__ u2, const float* __restrict__ g0,
                        const float* __restrict__ g1, const float* __restrict__ g2,
                        const int* __restrict__ species, const float* __restrict__ emb,
                        float* __restrict__ x0) {
    long t = (long)blockIdx.x * blockDim.x + threadIdx.x;
    if (t >= (long)N_ATOMS * KDIM) return;
    int k = (int)(t % KDIM);
    long n = t / KDIM;
    float f0v = f0[t];
    float feats0 = g0[t] * f0v + f0v;

    float G[9], F[9];
    float tr1 = 0.f;
#pragma unroll
    for (int p = 0; p < 9; ++p) {
        G[p] = g1[(n * 9 + p) * KDIM + k];
        F[p] = u1[(n * 9 + p) * KDIM + k];
    }
#pragma unroll
    for (int a = 0; a < 3; ++a)
#pragma unroll
        for (int b = 0; b < 3; ++b) tr1 += G[a * 3 + b] * F[b * 3 + a];
    tr1 += F[0] + F[4] + F[8];

    float tr2 = 0.f;
#pragma unroll
    for (int p = 0; p < 9; ++p) {
        G[p] = g2[(n * 9 + p) * KDIM + k];
        F[p] = u2[(n * 9 + p) * KDIM + k];
    }
#pragma unroll
    for (int a = 0; a < 3; ++a)
#pragma unroll
        for (int b = 0; b < 3; ++b) tr2 += G[a * 3 + b] * F[b * 3 + a];
    tr2 += F[0] + F[4] + F[8];

    const float i3 = 0.5773502691896258f;   // 1/sqrt(3) = U2[0] trace weight
    int sp = species[n];
    const float* e = emb + sp * N_CH;
    // embed: x[q] *= emb[q // 12]  (q in [0,384))
    x0[n * K0TOT + k]       = feats0      * e[k / 12];
    x0[n * K0TOT + 128 + k] = (tr1 * i3)  * e[(128 + k) / 12];
    x0[n * K0TOT + 256 + k] = (tr2 * i3)  * e[(256 + k) / 12];
}

// head GEMM (N x 384)(384 x 384): each wave computes a 16x128 stripe (8 tiles),
// then fuses bias + SiLU + dot with W_out, reducing the 8 column-tiles in
// registers so only ONE atomicAdd per (lane,row) reaches memory.
// grid = (N/16, 3), block = 32.
__global__ void k_head_wmma(const float* __restrict__ x0, const float* __restrict__ Wh,
                            const float* __restrict__ bh, const float* __restrict__ Wo,
                            float* __restrict__ out) {
    int m0 = blockIdx.x * 16;
    int n0 = blockIdx.y * 128;
    int lane = threadIdx.x;
    int half = lane >> 4;
    int l16  = lane & 15;
    const float* Arow = x0 + (long)(m0 + l16) * K0TOT;
    v8f acc[8] = {};
    for (int kk = 0; kk < K0TOT; kk += 4) {
        int ka = kk + 2 * half;
        v2f a;
        a.x = Arow[ka];
        a.y = Arow[ka + 1];
        const float* B0 = Wh + (long)ka * K0TOT + n0 + l16;
#pragma unroll
        for (int t = 0; t < 8; ++t) {
            v2f b;
            b.x = B0[t * 16];
            b.y = B0[K0TOT + t * 16];
            acc[t] = __builtin_amdgcn_wmma_f32_16x16x4_f32(false, a, false, b, (short)0,
                                                           acc[t], false, false);
        }
    }
    int rowbase = m0 + 8 * half;
    float partial[8];
#pragma unroll
    for (int i = 0; i < 8; ++i) partial[i] = 0.f;
#pragma unroll
    for (int t = 0; t < 8; ++t) {
        int col = n0 + t * 16 + l16;
        float bias = bh[col];
        float wo   = Wo[col];
#pragma unroll
        for (int i = 0; i < 8; ++i) {
            float v = acc[t][i] + bias;
            partial[i] += (v / (1.f + expf(-v))) * wo;   // SiLU * W_out
        }
    }
#pragma unroll
    for (int i = 0; i < 8; ++i) atomicAdd(&out[rowbase + i], partial[i]);
}

extern "C" void kernel_launch(void* const* d_in, const int* in_sizes, int n_in,
                              void* d_out, int out_size, void* d_ws, size_t ws_size,
                              hipStream_t stream) {
    (void)in_sizes; (void)n_in; (void)out_size; (void)ws_size;
    const float* positions = (const float*)d_in[0];
    const int*   species   = (const int*)d_in[1];
    const int*   pairs     = (const int*)d_in[2];
    const float* embeddings= (const float*)d_in[3];
    const float* W_rad     = (const float*)d_in[4];
    const float* W_cg0     = (const float*)d_in[5];
    const float* W_cg1     = (const float*)d_in[6];
    const float* W_cg2     = (const float*)d_in[7];
    const float* W_head    = (const float*)d_in[8];
    const float* b_head    = (const float*)d_in[9];
    const float* W_out     = (const float*)d_in[10];
    const float* b_out     = (const float*)d_in[11];
    float* out = (float*)d_out;

    // workspace layout (floats), total ~251 MB
    float* ws = (float*)d_ws;
    const long SZ_F0 = (long)N_ATOMS * KDIM;          // 1.28M
    const long SZ_F1 = (long)N_ATOMS * 3 * KDIM;      // 3.84M
    const long SZ_F2 = (long)N_ATOMS * 5 * KDIM;      // 6.40M
    const long SZ_U  = (long)N_ATOMS * 9 * KDIM;      // 11.52M
    float* f0 = ws;
    float* f1 = f0 + SZ_F0;
    float* f2 = f1 + SZ_F1;
    float* u1 = f2 + SZ_F2;
    float* u2 = u1 + SZ_U;
    float* g0 = u2 + SZ_U;
    float* g1 = g0 + SZ_F0;
    float* g2 = g1 + SZ_U;
    float* x0 = g2 + SZ_U;                            // N x 384

    // 1) zero the scatter targets (f0,f1,f2 are contiguous: 9*N*K floats)
    long nz = SZ_F0 + SZ_F1 + SZ_F2;                  // 11,520,000
    k_zero<<<(unsigned)((nz + 255) / 256), 256, 0, stream>>>(f0, nz);

    // 2) out[n] = b_out[0]
    k_init_out<<<(N_ATOMS + 255) / 256, 256, 0, stream>>>(out, b_out);

    // 3) pair messages + segment-sum via atomics
    k_pairs<<<N_PAIRS, KDIM, 0, stream>>>(positions, species, pairs, embeddings, W_rad,
                                          f0, f1, f2);

    // 4) uncouple (l=1,2): 10000*9*128 = 45000*256 exactly
    k_uncouple<<<45000, 256, 0, stream>>>(f1, f2, u1, u2);

    // 5-7) g_l = f_unc_l @ W_cg_l  (fp32 WMMA, 16x128 stripe per wave)
    k_gemm_wmma128<<<N_ATOMS / 16, 32, 0, stream>>>(f0, W_cg0, g0);
    k_gemm_wmma128<<<N_ATOMS * 9 / 16, 32, 0, stream>>>(u1, W_cg1, g1);
    k_gemm_wmma128<<<N_ATOMS * 9 / 16, 32, 0, stream>>>(u2, W_cg2, g2);

    // 8) feats + trace-couple + embed -> x0
    k_feats<<<(unsigned)(((long)N_ATOMS * KDIM + 255) / 256), 256, 0, stream>>>(
        f0, u1, u2, g0, g1, g2, species, embeddings, x0);

    // 9) head GEMM + SiLU + W_out reduction (fp32 WMMA, 16x128 stripe per wave)
    k_head_wmma<<<dim3(N_ATOMS / 16, K0TOT / 16 / 8), 32, 0, stream>>>(
        x0, W_head, b_head, W_out, out);
}